// UVRGCNLayer_47777216201145
// MI455X (gfx1250) — compile-verified
//
#include <hip/hip_runtime.h>

#define DD 128

typedef float v2f __attribute__((ext_vector_type(2)));
typedef float v8f __attribute__((ext_vector_type(8)));

__global__ void zero_f32(float* __restrict__ p, int n) {
    int i = blockIdx.x * blockDim.x + threadIdx.x;
    if (i < n) p[i] = 0.0f;
}

// WT[c*128+r] = W[r*128+c] for a 128x128 matrix (64 KB, trivial).
__global__ __launch_bounds__(256)
void transpose128(const float* __restrict__ W, float* __restrict__ WT) {
    int idx = blockIdx.x * blockDim.x + threadIdx.x;   // 0 .. 16383
    int r = idx >> 7;
    int c = idx & 127;
    WT[c * DD + r] = W[r * DD + c];
}

// O{0,1,2}[M,128] = A[M,128] @ W{0,1,2}  (weights given pre-transposed, [n][k]).
// Block = 256 threads = 8 wave32s. Each wave owns N-tile n0 = wave*16 and TWO
// M-tiles (m0, m0+16), so each k-step is 5 x global_load_b64 -> 6 x
// V_WMMA_F32_16X16X4_F32 with zero branches in the loop.
// A rows are clamped (not branched) for the grid tail; stores are guarded.
__global__ __launch_bounds__(256)
void gemm3_wmma_f32(const float* __restrict__ A, int M,
                    const float* __restrict__ W0T, float* __restrict__ O0,
                    const float* __restrict__ W1T, float* __restrict__ O1,
                    const float* __restrict__ W2T, float* __restrict__ O2) {
    const int lane = threadIdx.x & 31;
    const int wave = threadIdx.x >> 5;        // 0..7 -> N tile
    const int m0   = blockIdx.x << 5;         // 32 rows per block
    const int n0   = wave << 4;
    const int lm   = lane & 15;
    const int khi  = (lane >> 4) << 1;        // lanes 0-15: K0/K1, 16-31: K2/K3

    // A-fragment rows (clamped so tail blocks read valid memory)
    const int r0 = min(m0 + lm,      M - 1);
    const int r1 = min(m0 + 16 + lm, M - 1);
    const float* arow0 = A + (size_t)r0 * DD + khi;
    const float* arow1 = A + (size_t)r1 * DD + khi;

    // B-fragments from transposed weights: contiguous k-pair per lane
    const int col = n0 + lm;
    const float* bp0 = W0T + (size_t)col * DD + khi;
    const float* bp1 = W1T + (size_t)col * DD + khi;
    const float* bp2 = W2T + (size_t)col * DD + khi;

    v8f c00 = {}; v8f c01 = {}; v8f c02 = {};
    v8f c10 = {}; v8f c11 = {}; v8f c12 = {};

#pragma unroll 8
    for (int k0 = 0; k0 < DD; k0 += 4) {
        const v2f a0 = *(const v2f*)(arow0 + k0);
        const v2f a1 = *(const v2f*)(arow1 + k0);
        const v2f b0 = *(const v2f*)(bp0 + k0);
        const v2f b1 = *(const v2f*)(bp1 + k0);
        const v2f b2 = *(const v2f*)(bp2 + k0);
        c00 = __builtin_amdgcn_wmma_f32_16x16x4_f32(false, a0, false, b0, (short)0, c00, false, false);
        c10 = __builtin_amdgcn_wmma_f32_16x16x4_f32(false, a1, false, b0, (short)0, c10, false, false);
        c01 = __builtin_amdgcn_wmma_f32_16x16x4_f32(false, a0, false, b1, (short)0, c01, false, false);
        c11 = __builtin_amdgcn_wmma_f32_16x16x4_f32(false, a1, false, b1, (short)0, c11, false, false);
        c02 = __builtin_amdgcn_wmma_f32_16x16x4_f32(false, a0, false, b2, (short)0, c02, false, false);
        c12 = __builtin_amdgcn_wmma_f32_16x16x4_f32(false, a1, false, b2, (short)0, c12, false, false);
    }

    // C/D layout (ISA 7.12.2): VGPR r -> M = r (lanes 0-15) / r+8 (lanes 16-31)
    const int mBase0 = m0 + ((lane >> 4) << 3);
    const int mBase1 = mBase0 + 16;
#pragma unroll
    for (int r = 0; r < 8; ++r) {
        const int o0 = mBase0 + r;
        if (o0 < M) {
            O0[(size_t)o0 * DD + col] = c00[r];
            O1[(size_t)o0 * DD + col] = c01[r];
            O2[(size_t)o0 * DD + col] = c02[r];
        }
        const int o1 = mBase1 + r;
        if (o1 < M) {
            O0[(size_t)o1 * DD + col] = c10[r];
            O1[(size_t)o1 * DD + col] = c11[r];
            O2[(size_t)o1 * DD + col] = c12[r];
        }
    }
}

// Single-weight variant for relW = emb_rel @ Wn (M = 500, tail handled by
// clamp + guarded stores). One 16-row M-tile per block.
__global__ __launch_bounds__(256)
void gemm1_wmma_f32(const float* __restrict__ A, int M,
                    const float* __restrict__ WT, float* __restrict__ O) {
    const int lane = threadIdx.x & 31;
    const int wave = threadIdx.x >> 5;
    const int m0   = blockIdx.x << 4;
    const int n0   = wave << 4;
    const int lm   = lane & 15;
    const int khi  = (lane >> 4) << 1;

    const int r0 = min(m0 + lm, M - 1);
    const float* arow = A + (size_t)r0 * DD + khi;
    const int col = n0 + lm;
    const float* bp = WT + (size_t)col * DD + khi;

    v8f c = {};
#pragma unroll 8
    for (int k0 = 0; k0 < DD; k0 += 4) {
        const v2f a = *(const v2f*)(arow + k0);
        const v2f b = *(const v2f*)(bp + k0);
        c = __builtin_amdgcn_wmma_f32_16x16x4_f32(false, a, false, b, (short)0, c, false, false);
    }

    const int mBase = m0 + ((lane >> 4) << 3);
#pragma unroll
    for (int r = 0; r < 8; ++r) {
        const int orow = mBase + r;
        if (orow < M) O[(size_t)orow * DD + col] = c[r];
    }
}

// One wave32 per edge: 32 lanes x float4 = 128 floats.
// msg = hW[src] + relW[etype]; atomic scatter-sum into agg[dst] (L2-resident).
__global__ __launch_bounds__(256)
void edge_scatter(const int* __restrict__ src, const int* __restrict__ dst,
                  const int* __restrict__ rel,
                  const float* __restrict__ hW, const float* __restrict__ relW,
                  float* __restrict__ agg, float* __restrict__ indeg, int E) {
    const int e    = (blockIdx.x * blockDim.x + threadIdx.x) >> 5;
    const int lane = threadIdx.x & 31;
    if (e >= E) return;

    const int s = src[e];
    const int d = dst[e];
    const int r = rel[e];

    const float4 a = ((const float4*)(hW   + (size_t)s * DD))[lane];
    const float4 b = ((const float4*)(relW + (size_t)r * DD))[lane];

    float* o = agg + (size_t)d * DD + lane * 4;
    atomicAdd(o + 0, a.x + b.x);
    atomicAdd(o + 1, a.y + b.y);
    atomicAdd(o + 2, a.z + b.z);
    atomicAdd(o + 3, a.w + b.w);
    if (lane == 0) atomicAdd(indeg + d, 1.0f);
}

// out = relu(agg*norm + (indeg>0 ? hL : hE)), float4 per thread, in-place.
__global__ __launch_bounds__(256)
void finalize_relu(float* __restrict__ out, const float* __restrict__ norm,
                   const float* __restrict__ hL, const float* __restrict__ hE,
                   const float* __restrict__ indeg, int N) {
    const int t = blockIdx.x * blockDim.x + threadIdx.x;
    if (t >= N * (DD / 4)) return;
    const int node = t >> 5;
    const int off  = t & 31;

    const float nn = norm[node];
    const float* lsrc = (indeg[node] > 0.0f) ? hL : hE;
    const float4 l = ((const float4*)(lsrc + (size_t)node * DD))[off];

    float4* op = (float4*)(out + (size_t)node * DD) + off;
    float4 a = *op;
    a.x = fmaxf(a.x * nn + l.x, 0.0f);
    a.y = fmaxf(a.y * nn + l.y, 0.0f);
    a.z = fmaxf(a.z * nn + l.z, 0.0f);
    a.w = fmaxf(a.w * nn + l.w, 0.0f);
    *op = a;
}

extern "C" void kernel_launch(void* const* d_in, const int* in_sizes, int n_in,
                              void* d_out, int out_size, void* d_ws, size_t ws_size,
                              hipStream_t stream) {
    const float* h        = (const float*)d_in[0];
    const float* norm     = (const float*)d_in[1];
    const float* emb_rel  = (const float*)d_in[2];
    const float* Wn       = (const float*)d_in[3];
    const float* Wloop    = (const float*)d_in[4];
    const float* Wevolve  = (const float*)d_in[5];
    const int*   src      = (const int*)d_in[6];
    const int*   dst      = (const int*)d_in[7];
    const int*   ety      = (const int*)d_in[8];
    float*       out      = (float*)d_out;

    const int N = in_sizes[1];          // norm is [N,1]
    const int R = in_sizes[2] / DD;     // emb_rel is [R,D]
    const int E = in_sizes[6];

    // Workspace layout
    float* hW    = (float*)d_ws;
    float* hL    = hW + (size_t)N * DD;
    float* hE    = hL + (size_t)N * DD;
    float* relW  = hE + (size_t)N * DD;
    float* indeg = relW + (size_t)((R * DD + 63) & ~63);
    float* WnT   = indeg + (size_t)((N + 63) & ~63);
    float* WlT   = WnT + DD * DD;
    float* WeT   = WlT + DD * DD;

    // 1) zero the scatter accumulator (d_out) and the in-degree counters
    const int totO = N * DD;
    zero_f32<<<(totO + 255) / 256, 256, 0, stream>>>(out, totO);
    zero_f32<<<(N + 255) / 256, 256, 0, stream>>>(indeg, N);

    // 2) transpose the three weights so B-fragments are contiguous b64 loads
    transpose128<<<(DD * DD) / 256, 256, 0, stream>>>(Wn, WnT);
    transpose128<<<(DD * DD) / 256, 256, 0, stream>>>(Wloop, WlT);
    transpose128<<<(DD * DD) / 256, 256, 0, stream>>>(Wevolve, WeT);

    // 3) hW = h@Wn, hL = h@Wloop, hE = h@Wevolve (shared A-fragments, WMMA f32)
    gemm3_wmma_f32<<<(N + 31) / 32, 256, 0, stream>>>(h, N, WnT, hW,
                                                      WlT, hL, WeT, hE);

    // 4) relW = emb_rel @ Wn (M=500, tail tile clamped+guarded)
    gemm1_wmma_f32<<<(R + 15) / 16, 256, 0, stream>>>(emb_rel, R, WnT, relW);

    // 5) per-edge gather + atomic scatter-sum (one wave32 per edge)
    edge_scatter<<<(E * 32 + 255) / 256, 256, 0, stream>>>(src, dst, ety,
                                                           hW, relW, out,
                                                           indeg, E);

    // 6) out = relu(out*norm + (deg>0 ? hL : hE))
    finalize_relu<<<(N * (DD / 4) + 255) / 256, 256, 0, stream>>>(out, norm,
                                                                  hL, hE,
                                                                  indeg, N);
}